// CAMELCore_14070312862149
// MI455X (gfx1250) — compile-verified
//
#include <hip/hip_runtime.h>
#include <hip/hip_bf16.h>
#include <math.h>

#define BB 16
#define TT 96
#define NN 716
#define DD 64
#define MM 1196
#define KK 8
#define HORZ 12
#define HIDN 32
#define NHD 4
#define TCONV 97
#define ND (NN*DD)   // 45824

typedef __attribute__((ext_vector_type(2))) float v2f;
typedef __attribute__((ext_vector_type(8))) float v8f;

__device__ __forceinline__ float gelu_f(float x){
  return 0.5f * x * (1.0f + erff(x * 0.70710678118654752440f));
}
__device__ __forceinline__ float softplus_f(float x){
  return (x > 20.0f) ? x : log1pf(expf(x));
}
__device__ __forceinline__ float wsum32(float v){
  #pragma unroll
  for (int o = 16; o > 0; o >>= 1) v += __shfl_xor(v, o, 32);
  return v;
}
__device__ __forceinline__ float wsum16(float v){   // reduce within 16-lane halves
  #pragma unroll
  for (int o = 8; o > 0; o >>= 1) v += __shfl_xor(v, o, 32);
  return v;
}

// ---------------------------------------------------------------------------
// 1. depthwise conv (width 12, pad 6) + GELU -> h1 stored TIME-MAJOR (B, 97, N)
//    so the pointwise GEMM's B fragments are contiguous b64 loads.
__global__ void k_dwconv(const float* __restrict__ x, const float* __restrict__ w,
                         const float* __restrict__ bias, float* __restrict__ h1){
  int idx = blockIdx.x * blockDim.x + threadIdx.x;
  if (idx >= BB * TCONV * NN) return;
  int n  = idx % NN;
  int tc = (idx / NN) % TCONV;
  int b  = idx / (NN * TCONV);
  float s = bias[n];
  #pragma unroll
  for (int k = 0; k < 12; ++k){
    int ti = tc - 6 + k;
    if (ti >= 0 && ti < TT) s += x[((size_t)b * TT + ti) * NN + n] * w[n * 12 + k];
  }
  h1[idx] = gelu_f(s);   // h1[(b*97 + tc)*716 + n]
}

// ---------------------------------------------------------------------------
// 2. pointwise GEMM (45824x716)@(716x97) per batch, fused GELU + time-sum.
//    blockIdx.z splits the 7 time tiles: split 0 -> tiles {0,1,2,3},
//    split 1 -> tiles {3,4,5,6} with tile 3 masked out (counted once).
//    Only 4 v8f accumulators per wave -> no register spills even with the
//    compiler's x2 unroll + pipelining. Each split stores a private partial
//    sum (deterministic, no atomics); the LN kernel combines them and /97.
__global__ void k_pw_gemm_mean(const float* __restrict__ h1, const float* __restrict__ pw_w,
                               const float* __restrict__ pw_b, float* __restrict__ qp){
  int b     = blockIdx.y;
  int o0    = blockIdx.x * 16;
  int split = blockIdx.z;
  int tbase = split * 3;
  int lane  = threadIdx.x;
  int l16 = lane & 15;
  int hl  = lane >> 4;
  v8f zero = {};
  v8f acc[4];
  #pragma unroll
  for (int i = 0; i < 4; ++i) acc[i] = zero;

  // per-lane clamped time index per tile (uniform across k loop)
  int tcl[4];
  #pragma unroll
  for (int i = 0; i < 4; ++i){
    int t = (tbase + i) * 16 + l16;
    tcl[i] = (t < TCONV) ? t : (TCONV - 1);
  }

  const float* Arow = pw_w + (size_t)(o0 + l16) * NN + 2 * hl; // A: 16x4 f32 frag
  const float* Bb   = h1 + (size_t)b * TCONV * NN + 2 * hl;    // time-major

  for (int c = 0; c < NN; c += 4){
    if ((c & 127) == 0){
      __builtin_prefetch(Arow + c + 128, 0, 0);   // global_prefetch_b8
    }
    v2f a = *(const v2f*)(Arow + c);
    #pragma unroll
    for (int i = 0; i < 4; ++i){
      v2f bv = *(const v2f*)(Bb + (size_t)tcl[i] * NN + c);
      acc[i] = __builtin_amdgcn_wmma_f32_16x16x4_f32(false, a, false, bv,
                                                     (short)0, acc[i], false, false);
    }
  }
  #pragma unroll
  for (int r = 0; r < 8; ++r){
    int orow = o0 + r + 8 * hl;
    float pb = pw_b[orow];
    float s = 0.0f;
    #pragma unroll
    for (int i = 0; i < 4; ++i){
      int t = (tbase + i) * 16 + l16;
      bool inc = (t < TCONV) && !(split == 1 && i == 0);
      float v = acc[i][r] + pb;
      s += inc ? gelu_f(v) : 0.0f;
    }
    s = wsum16(s);
    if (l16 == 0)
      qp[(size_t)split * BB * ND + (size_t)b * ND + orow] = s;
  }
}

// ---------------------------------------------------------------------------
// 3. generic WMMA GEMM: out[r,o] = act( sum_k A[r,k] * W[o,k] + bias[o] )
//    Optional row indirection: A row rr -> rmap[rr/rows_per_idx]*rows_per_idx + rr%rows_per_idx
__global__ void k_gemm_wmma(const float* __restrict__ A, int lda,
                            const float* __restrict__ W,
                            const float* __restrict__ bias,
                            float* __restrict__ outp,
                            int R, int Kdim, int Odim, int act,
                            const int* __restrict__ rmap, int rows_per_idx){
  int r0 = blockIdx.x * 16;
  int o0 = blockIdx.y * 16;
  int lane = threadIdx.x;
  int l16 = lane & 15;
  int hl  = lane >> 4;
  int rrow = r0 + l16; int rc = (rrow < R) ? rrow : (R - 1);
  size_t abase;
  if (rmap){
    int grp = rc / rows_per_idx;
    int off = rc - grp * rows_per_idx;
    abase = ((size_t)rmap[grp] * rows_per_idx + off) * (size_t)lda;
  } else {
    abase = (size_t)rc * (size_t)lda;
  }
  int ocol = o0 + l16; int oc = (ocol < Odim) ? ocol : (Odim - 1);
  const float* Ap = A + abase + 2 * hl;
  const float* Wp = W + (size_t)oc * Kdim + 2 * hl;
  v8f acc = {};
  for (int k = 0; k < Kdim; k += 4){
    if ((k & 127) == 0){
      __builtin_prefetch(Ap + k + 128, 0, 0);   // global_prefetch_b8
      __builtin_prefetch(Wp + k + 128, 0, 0);
    }
    v2f a = *(const v2f*)(Ap + k);
    v2f w = *(const v2f*)(Wp + k);
    acc = __builtin_amdgcn_wmma_f32_16x16x4_f32(false, a, false, w,
                                                (short)0, acc, false, false);
  }
  float bv = 0.0f;
  if (bias && ocol < Odim) bv = bias[ocol];
  #pragma unroll
  for (int r = 0; r < 8; ++r){
    int m = r0 + r + 8 * hl;
    if (m < R && ocol < Odim){
      float v = acc[r] + bv;
      if (act) v = gelu_f(v);
      outp[(size_t)m * Odim + ocol] = v;
    }
  }
}

// ---------------------------------------------------------------------------
// 4. LayerNorm over rows of 64.
//    x = (in[row] + in2[row]) * scale   (in2 optional, for combining partials)
//    Optionally writes per-row sum-of-squares of the normalized output
//    (plain store -> deterministic; reduced per-batch by k_bred).
__global__ void k_ln_rows(const float* __restrict__ in, const float* __restrict__ in2,
                          float scale, float* __restrict__ outp,
                          const float* __restrict__ g, const float* __restrict__ b,
                          float* __restrict__ rowss){
  int row = blockIdx.x;
  int lane = threadIdx.x;
  const float* rp = in + (size_t)row * DD;
  float x0 = rp[2 * lane], x1 = rp[2 * lane + 1];
  if (in2){
    const float* rp2 = in2 + (size_t)row * DD;
    x0 += rp2[2 * lane]; x1 += rp2[2 * lane + 1];
  }
  x0 *= scale; x1 *= scale;
  float s  = wsum32(x0 + x1);
  float s2 = wsum32(x0 * x0 + x1 * x1);
  float mu  = s * (1.0f / 64.0f);
  float var = s2 * (1.0f / 64.0f) - mu * mu;
  float inv = rsqrtf(var + 1e-5f);
  float y0 = (x0 - mu) * inv * g[2 * lane]     + b[2 * lane];
  float y1 = (x1 - mu) * inv * g[2 * lane + 1] + b[2 * lane + 1];
  float* op = outp + (size_t)row * DD;
  op[2 * lane] = y0; op[2 * lane + 1] = y1;
  if (rowss){
    float qq = wsum32(y0 * y0 + y1 * y1);
    if (lane == 0) rowss[row] = qq;
  }
}

// per-batch deterministic reduction of per-row sum-of-squares
__global__ void k_bred(const float* __restrict__ rowss, float* __restrict__ qss){
  int b = blockIdx.x;
  int lane = threadIdx.x;
  float s = 0.0f;
  for (int i = lane; i < NN; i += 32) s += rowss[(size_t)b * NN + i];
  s = wsum32(s);
  if (lane == 0) qss[b] = s;
}

// ---------------------------------------------------------------------------
// 5. per-memory-row L2 norm (over 45824 elems)
__global__ void k_mem_norm(const float* __restrict__ mem, float* __restrict__ mn){
  __shared__ float red[256];
  int m = blockIdx.x;
  const float* rp = mem + (size_t)m * ND;
  float s = 0.0f;
  for (int i = threadIdx.x; i < ND; i += 256){ float v = rp[i]; s += v * v; }
  red[threadIdx.x] = s; __syncthreads();
  for (int o = 128; o > 0; o >>= 1){
    if (threadIdx.x < o) red[threadIdx.x] += red[threadIdx.x + o];
    __syncthreads();
  }
  if (threadIdx.x == 0) mn[m] = fmaxf(sqrtf(red[0]), 1e-12f);
}

// ---------------------------------------------------------------------------
// 6. sim normalization + season mask + diversity scaling
__global__ void k_sim_adjust(float* __restrict__ sim, const float* __restrict__ qss,
                             const float* __restrict__ mn,
                             const int* __restrict__ season_q, const int* __restrict__ mem_season,
                             const float* __restrict__ year_q, const float* __restrict__ mem_year){
  int t = blockIdx.x * blockDim.x + threadIdx.x;
  if (t >= BB * MM) return;
  int b = t / MM, m = t % MM;
  float qn = fmaxf(sqrtf(qss[b]), 1e-12f);
  float s = sim[t] / (qn * mn[m]);
  if (season_q[b] != mem_season[m]) s = -10000.0f;
  float dy = fabsf(year_q[b] - mem_year[m]);
  float div = 1.0f - expf(-dy * 0.5f);   // TAU_TIME = 2
  sim[t] = s * (0.5f + 0.5f * div);
}

// ---------------------------------------------------------------------------
// 7. top-8 per batch (one wave per batch, 8 argmax passes)
__global__ void k_topk(const float* __restrict__ sim, int* __restrict__ idx_out){
  int b = blockIdx.x;
  int lane = threadIdx.x;
  const float* s = sim + (size_t)b * MM;
  int chosen[KK];
  for (int p = 0; p < KK; ++p){
    float bestv = -3.4e38f; int besti = 0x7fffffff;
    for (int m = lane; m < MM; m += 32){
      bool used = false;
      for (int j = 0; j < p; ++j) used |= (chosen[j] == m);
      if (!used){
        float v = s[m];
        if (v > bestv || (v == bestv && m < besti)){ bestv = v; besti = m; }
      }
    }
    #pragma unroll
    for (int o = 16; o > 0; o >>= 1){
      float ov = __shfl_xor(bestv, o, 32);
      int   oi = __shfl_xor(besti, o, 32);
      if (ov > bestv || (ov == bestv && oi < besti)){ bestv = ov; besti = oi; }
    }
    chosen[p] = besti;
    if (lane == 0) idx_out[b * KK + p] = besti;
  }
}

// ---------------------------------------------------------------------------
// 8. attention: wave per (b,n,h); lanes hold hd=16 elems (duplicated in halves)
__global__ void k_attn(const float* __restrict__ qh, const float* __restrict__ kh,
                       const float* __restrict__ vh, float* __restrict__ o){
  int idx = blockIdx.x;
  int h = idx & 3;
  int n = (idx >> 2) % NN;
  int b = idx / (NHD * NN);
  int lane = threadIdx.x;
  int e = lane & 15;
  size_t qoff = ((size_t)(b * NN + n)) * DD + h * 16 + e;
  float qv = qh[qoff];
  float sc[KK];
  #pragma unroll
  for (int k = 0; k < KK; ++k){
    size_t koff = (((size_t)(b * KK + k) * NN) + n) * DD + h * 16 + e;
    float p = qv * kh[koff];
    p = wsum16(p);
    sc[k] = p * 0.25f;   // 1/sqrt(hd=16)
  }
  float mx = sc[0];
  #pragma unroll
  for (int k = 1; k < KK; ++k) mx = fmaxf(mx, sc[k]);
  float den = 0.0f; float w[KK];
  #pragma unroll
  for (int k = 0; k < KK; ++k){ w[k] = expf(sc[k] - mx); den += w[k]; }
  float inv = 1.0f / den;
  float ov = 0.0f;
  #pragma unroll
  for (int k = 0; k < KK; ++k){
    size_t voff = (((size_t)(b * KK + k) * NN) + n) * DD + h * 16 + e;
    ov += w[k] * inv * vh[voff];
  }
  if (lane < 16) o[qoff] = ov;
}

// ---------------------------------------------------------------------------
// 9. build cat = [h_cem, h_lde, x_enc, gap_embed[0]]  (B,N,256)
__global__ void k_cat(const float* __restrict__ h_cem, const float* __restrict__ h_lde,
                      const float* __restrict__ x, const float* __restrict__ xp_w,
                      const float* __restrict__ xp_b, const float* __restrict__ gap,
                      float* __restrict__ cat){
  int t = blockIdx.x * blockDim.x + threadIdx.x;
  if (t >= BB * NN * 256) return;
  int j = t & 255; int bn = t >> 8;
  int b = bn / NN; int n = bn % NN;
  float v;
  if (j < 64)        v = h_cem[(size_t)bn * 64 + j];
  else if (j < 128)  v = h_lde[(size_t)bn * 64 + (j - 64)];
  else if (j < 192){ int e = j - 128;
                     v = x[((size_t)b * TT + (TT - 1)) * NN + n] * xp_w[e] + xp_b[e]; }
  else               v = gap[j - 192];
  cat[t] = v;
}

// 10. gate logits (3 x 128 dot) + softmax
__global__ void k_gates(const float* __restrict__ g1o, const float* __restrict__ g2_w,
                        const float* __restrict__ g2_b, float* __restrict__ gates){
  int bn = blockIdx.x * blockDim.x + threadIdx.x;
  if (bn >= BB * NN) return;
  const float* gp = g1o + (size_t)bn * 128;
  float lg[3];
  #pragma unroll
  for (int c = 0; c < 3; ++c){
    const float* wp = g2_w + c * 128;
    float s = g2_b[c];
    for (int k = 0; k < 128; ++k) s += gp[k] * wp[k];
    lg[c] = s;
  }
  float mx = fmaxf(lg[0], fmaxf(lg[1], lg[2]));
  float e0 = expf(lg[0] - mx), e1 = expf(lg[1] - mx), e2 = expf(lg[2] - mx);
  float inv = 1.0f / (e0 + e1 + e2);
  gates[bn * 3 + 0] = e0 * inv;
  gates[bn * 3 + 1] = e1 * inv;
  gates[bn * 3 + 2] = e2 * inv;
}

// 11. z = g0*h_cem + g1*h_lde + g2*x_enc
__global__ void k_zmix(const float* __restrict__ gates, const float* __restrict__ h_cem,
                       const float* __restrict__ h_lde, const float* __restrict__ x,
                       const float* __restrict__ xp_w, const float* __restrict__ xp_b,
                       float* __restrict__ z){
  int t = blockIdx.x * blockDim.x + threadIdx.x;
  if (t >= BB * NN * DD) return;
  int e = t & 63; int bn = t >> 6;
  int b = bn / NN; int n = bn % NN;
  float xe = x[((size_t)b * TT + (TT - 1)) * NN + n] * xp_w[e] + xp_b[e];
  z[t] = gates[bn * 3 + 0] * h_cem[t] + gates[bn * 3 + 1] * h_lde[t] + gates[bn * 3 + 2] * xe;
}

// 12. sigma head: (B,N,12) = softplus(u1out @ u2^T + b) + 1e-6
__global__ void k_sigma(const float* __restrict__ u1o, const float* __restrict__ u2_w,
                        const float* __restrict__ u2_b, float* __restrict__ sig){
  int t = blockIdx.x * blockDim.x + threadIdx.x;
  if (t >= BB * NN * HORZ) return;
  int hor = t % HORZ; int bn = t / HORZ;
  const float* up = u1o + (size_t)bn * HIDN;
  const float* wp = u2_w + (size_t)hor * HIDN;
  float s = u2_b[hor];
  #pragma unroll
  for (int k = 0; k < HIDN; ++k) s += up[k] * wp[k];
  sig[t] = softplus_f(s) + 1e-6f;
}

// 13. delta head
__global__ void k_delta(const float* __restrict__ z, const float* __restrict__ dh_w,
                        const float* __restrict__ dh_b, float* __restrict__ delta){
  int bn = blockIdx.x * blockDim.x + threadIdx.x;
  if (bn >= BB * NN) return;
  const float* zp = z + (size_t)bn * DD;
  float s = dh_b[0];
  for (int k = 0; k < DD; ++k) s += zp[k] * dh_w[k];
  delta[bn] = s;
}

// 14. z_out = x_scalar + delta broadcast over time
__global__ void k_zout(const float* __restrict__ x, const float* __restrict__ delta,
                       float* __restrict__ outp){
  int t = blockIdx.x * blockDim.x + threadIdx.x;
  if (t >= BB * TT * NN) return;
  int n = t % NN; int b = t / (TT * NN);
  outp[t] = x[t] + delta[b * NN + n];
}

// ---------------------------------------------------------------------------
extern "C" void kernel_launch(void* const* d_in, const int* in_sizes, int n_in,
                              void* d_out, int out_size, void* d_ws, size_t ws_size,
                              hipStream_t stream){
  (void)in_sizes; (void)n_in; (void)out_size; (void)ws_size;
  const float* x         = (const float*)d_in[0];
  const int*   season_q  = (const int*)  d_in[1];
  const float* year_q    = (const float*)d_in[2];
  const float* h_lde     = (const float*)d_in[3];
  const float* dw_w      = (const float*)d_in[4];
  const float* dw_b      = (const float*)d_in[5];
  const float* pw_w      = (const float*)d_in[6];
  const float* pw_b      = (const float*)d_in[7];
  const float* enc_g     = (const float*)d_in[8];
  const float* enc_b     = (const float*)d_in[9];
  const float* mem_bank  = (const float*)d_in[10];
  const int*   mem_season= (const int*)  d_in[11];
  const float* mem_year  = (const float*)d_in[12];
  const float* in_proj_w = (const float*)d_in[13];
  const float* in_proj_b = (const float*)d_in[14];
  const float* attn_ow   = (const float*)d_in[15];
  const float* attn_ob   = (const float*)d_in[16];
  const float* proj_w    = (const float*)d_in[17];
  const float* proj_b    = (const float*)d_in[18];
  const float* gap_embed = (const float*)d_in[19];
  const float* xp_w      = (const float*)d_in[20];
  const float* xp_b      = (const float*)d_in[21];
  const float* g1_w      = (const float*)d_in[22];
  const float* g1_b      = (const float*)d_in[23];
  const float* g2_w      = (const float*)d_in[24];
  const float* g2_b      = (const float*)d_in[25];
  const float* op_w      = (const float*)d_in[26];
  const float* op_b      = (const float*)d_in[27];
  const float* dh_w      = (const float*)d_in[28];
  const float* dh_b      = (const float*)d_in[29];
  const float* fln_g     = (const float*)d_in[30];
  const float* fln_b     = (const float*)d_in[31];
  const float* u1_w      = (const float*)d_in[32];
  const float* u1_b      = (const float*)d_in[33];
  const float* u2_w      = (const float*)d_in[34];
  const float* u2_b      = (const float*)d_in[35];
  float* out = (float*)d_out;
  float* ws  = (float*)d_ws;

  // workspace layout (floats) — all offsets even, so v2f (8B) loads stay aligned
  constexpr size_t N_H1  = (size_t)BB * TCONV * NN;     // 1,111,232 (time-major)
  constexpr size_t N_Q   = (size_t)BB * ND;             //   733,184
  constexpr size_t N_KH  = (size_t)BB * KK * NN * DD;   // 5,865,472
  constexpr size_t N_CAT = (size_t)BB * NN * 256;
  constexpr size_t N_G1  = (size_t)BB * NN * 128;
  constexpr size_t N_U1  = (size_t)BB * NN * HIDN;
  size_t F_H1   = 0;
  size_t F_Q    = F_H1 + N_H1;
  size_t F_QSS  = F_Q + N_Q;
  size_t F_MN   = F_QSS + BB;
  size_t F_SIM  = F_MN + MM;
  size_t F_IDX  = F_SIM + (size_t)BB * MM;
  size_t F_QH   = F_IDX + BB * KK;
  size_t F_KH   = F_QH + N_Q;
  size_t F_VH   = F_KH + N_KH;
  size_t F_O    = F_VH + N_KH;
  size_t F_TMP  = F_O + N_Q;
  size_t F_HCEM = F_TMP + N_Q;
  size_t F_CAT  = F_HCEM + N_Q;
  size_t F_G1   = F_CAT + N_CAT;
  size_t F_GATE = F_G1 + N_G1;
  size_t F_Z    = F_GATE + (size_t)BB * NN * 3;
  size_t F_ZO   = F_Z + N_Q;
  size_t F_U1   = F_ZO + N_Q;
  size_t F_DEL  = F_U1 + N_U1;
  size_t F_QP   = F_DEL + (size_t)BB * NN;   // 2 partial-sum buffers (split 0/1)
  size_t F_RSS  = F_QP + 2 * N_Q;            // per-row sum-of-squares

  // 1. depthwise conv + gelu (time-major output)
  {
    int n = BB * TCONV * NN;
    k_dwconv<<<(n + 255) / 256, 256, 0, stream>>>(x, dw_w, dw_b, ws + F_H1);
  }
  // 2. pointwise WMMA GEMM + gelu + time-sum (two deterministic partials)
  k_pw_gemm_mean<<<dim3(ND / 16, BB, 2), 32, 0, stream>>>(ws + F_H1, pw_w, pw_b, ws + F_QP);
  // 3. encoder LN: combines partials, /97, writes q + per-row ssq; then per-batch reduce
  k_ln_rows<<<BB * NN, 32, 0, stream>>>(ws + F_QP, ws + F_QP + N_Q, 1.0f / 97.0f,
                                        ws + F_Q, enc_g, enc_b, ws + F_RSS);
  k_bred<<<BB, 32, 0, stream>>>(ws + F_RSS, ws + F_QSS);
  // 4. memory norms + raw sim (16 x 1196 over K=45824 via WMMA)
  k_mem_norm<<<MM, 256, 0, stream>>>(mem_bank, ws + F_MN);
  k_gemm_wmma<<<dim3(1, (MM + 15) / 16), 32, 0, stream>>>(
      ws + F_Q, ND, mem_bank, nullptr, ws + F_SIM, BB, ND, MM, 0, nullptr, 0);
  k_sim_adjust<<<(BB * MM + 255) / 256, 256, 0, stream>>>(
      ws + F_SIM, ws + F_QSS, ws + F_MN, season_q, mem_season, year_q, mem_year);
  // 5. top-8 per batch
  k_topk<<<BB, 32, 0, stream>>>(ws + F_SIM, (int*)(ws + F_IDX));
  // 6. q/k/v projections (k,v gather rows from memory_bank via topk indices)
  k_gemm_wmma<<<dim3(BB * NN / 16, 4), 32, 0, stream>>>(
      ws + F_Q, DD, in_proj_w, in_proj_b, ws + F_QH, BB * NN, DD, DD, 0, nullptr, 0);
  k_gemm_wmma<<<dim3(BB * KK * NN / 16, 4), 32, 0, stream>>>(
      mem_bank, DD, in_proj_w + DD * DD, in_proj_b + DD, ws + F_KH,
      BB * KK * NN, DD, DD, 0, (const int*)(ws + F_IDX), NN);
  k_gemm_wmma<<<dim3(BB * KK * NN / 16, 4), 32, 0, stream>>>(
      mem_bank, DD, in_proj_w + 2 * DD * DD, in_proj_b + 2 * DD, ws + F_VH,
      BB * KK * NN, DD, DD, 0, (const int*)(ws + F_IDX), NN);
  // 7. attention over K=8
  k_attn<<<BB * NN * NHD, 32, 0, stream>>>(ws + F_QH, ws + F_KH, ws + F_VH, ws + F_O);
  // 8. output projections
  k_gemm_wmma<<<dim3(BB * NN / 16, 4), 32, 0, stream>>>(
      ws + F_O, DD, attn_ow, attn_ob, ws + F_TMP, BB * NN, DD, DD, 0, nullptr, 0);
  k_gemm_wmma<<<dim3(BB * NN / 16, 4), 32, 0, stream>>>(
      ws + F_TMP, DD, proj_w, proj_b, ws + F_HCEM, BB * NN, DD, DD, 0, nullptr, 0);
  // 9. gating
  k_cat<<<(BB * NN * 256 + 255) / 256, 256, 0, stream>>>(
      ws + F_HCEM, h_lde, x, xp_w, xp_b, gap_embed, ws + F_CAT);
  k_gemm_wmma<<<dim3(BB * NN / 16, 8), 32, 0, stream>>>(
      ws + F_CAT, 256, g1_w, g1_b, ws + F_G1, BB * NN, 256, 128, 1, nullptr, 0);
  k_gates<<<(BB * NN + 255) / 256, 256, 0, stream>>>(ws + F_G1, g2_w, g2_b, ws + F_GATE);
  k_zmix<<<(BB * NN * DD + 255) / 256, 256, 0, stream>>>(
      ws + F_GATE, ws + F_HCEM, h_lde, x, xp_w, xp_b, ws + F_Z);
  // 10. output proj + final LN (in place, no partial combine)
  k_gemm_wmma<<<dim3(BB * NN / 16, 4), 32, 0, stream>>>(
      ws + F_Z, DD, op_w, op_b, ws + F_ZO, BB * NN, DD, DD, 0, nullptr, 0);
  k_ln_rows<<<BB * NN, 32, 0, stream>>>(ws + F_ZO, nullptr, 1.0f,
                                        ws + F_ZO, fln_g, fln_b, nullptr);
  // 11. sigma head
  k_gemm_wmma<<<dim3(BB * NN / 16, 2), 32, 0, stream>>>(
      ws + F_ZO, DD, u1_w, u1_b, ws + F_U1, BB * NN, DD, HIDN, 1, nullptr, 0);
  k_sigma<<<(BB * NN * HORZ + 255) / 256, 256, 0, stream>>>(
      ws + F_U1, u2_w, u2_b, out + (size_t)BB * TT * NN);
  // 12. delta + z_out
  k_delta<<<(BB * NN + 255) / 256, 256, 0, stream>>>(ws + F_ZO, dh_w, dh_b, ws + F_DEL);
  k_zout<<<(BB * TT * NN + 255) / 256, 256, 0, stream>>>(x, ws + F_DEL, out);
}